// RKA_MultiheadAttention_22247930593750
// MI455X (gfx1250) — compile-verified
//
#include <hip/hip_runtime.h>
#include <hip/hip_bf16.h>
#include <math.h>

typedef float v2f __attribute__((ext_vector_type(2)));
typedef float v8f __attribute__((ext_vector_type(8)));

#define L_SEQ   2048
#define B_HEADS 32
#define HD      64
#define BH_ELEMS 4194304ull /* 32*2048*64 */

__device__ __forceinline__ v8f wmma_f32(v2f a, v2f b, v8f c) {
  // V_WMMA_F32_16X16X4_F32 : D = A(16x4) * B(4x16) + C(16x16), all fp32
  return __builtin_amdgcn_wmma_f32_16x16x4_f32(false, a, false, b, (short)0, c,
                                               false, false);
}

// ---- CDNA5 async global->LDS (bypasses VGPRs, tracked by ASYNCcnt) --------
typedef __attribute__((address_space(3))) void lds_void_t;

__device__ __forceinline__ unsigned lds_addr_of(void* p) {
  return (unsigned)(unsigned long long)(lds_void_t*)p;
}

__device__ __forceinline__ void async_copy_b128(unsigned lds_off,
                                                const float* gaddr) {
  // per-lane: copies 16 bytes from global address gaddr to LDS address lds_off
  asm volatile("global_load_async_to_lds_b128 %0, %1, off"
               :: "v"(lds_off), "v"(gaddr)
               : "memory");
}

#define WAIT_ASYNC(n) asm volatile("s_wait_asynccnt %0" :: "n"(n) : "memory")

// ---------------------------------------------------------------------------
// Generic C = X(MxK) @ W(NcxK)^T + bias, fp32 WMMA, 64x64 tile per 256-thr
// block, double-buffered async-LDS pipeline over K.
// mode 0: row-major store to out. mode 1: scatter into q/k/v head layout.
// ---------------------------------------------------------------------------
#define LSTR 36
__global__ __launch_bounds__(256)
void gemm_xwT(const float* __restrict__ X, const float* __restrict__ W,
              const float* __restrict__ bias, float* __restrict__ out,
              float* __restrict__ qkv_base, int M, int Nc, int K, int mode)
{
  __shared__ __align__(16) float As[2][64 * LSTR];
  __shared__ __align__(16) float Ws[2][64 * LSTR];
  const int tid  = threadIdx.x;
  const int lane = tid & 31, w = tid >> 5;
  const int hf   = lane >> 4, lid = lane & 15;
  const int m0   = blockIdx.y * 64, n0 = blockIdx.x * 64;
  const int r16  = (w & 3) * 16;
  const int c0   = (w >> 2) * 32;

  v8f acc0 = {}; v8f acc1 = {};

  const int NT = K >> 5;  // K/32 tiles

  // prologue: async-fetch tile 0 into buffer 0
  {
    for (int f = tid; f < 512; f += 256) {
      const int rr = f >> 3, kq = (f & 7) * 4;
      async_copy_b128(lds_addr_of(&As[0][rr * LSTR + kq]),
                      X + (size_t)(m0 + rr) * K + kq);
      async_copy_b128(lds_addr_of(&Ws[0][rr * LSTR + kq]),
                      W + (size_t)(n0 + rr) * K + kq);
    }
  }

  for (int it = 0; it < NT; ++it) {
    const int cur = it & 1;
    if (it + 1 < NT) {
      const int k0n = (it + 1) << 5;
      for (int f = tid; f < 512; f += 256) {
        const int rr = f >> 3, kq = (f & 7) * 4;
        async_copy_b128(lds_addr_of(&As[cur ^ 1][rr * LSTR + kq]),
                        X + (size_t)(m0 + rr) * K + k0n + kq);
        async_copy_b128(lds_addr_of(&Ws[cur ^ 1][rr * LSTR + kq]),
                        W + (size_t)(n0 + rr) * K + k0n + kq);
      }
      WAIT_ASYNC(4);   // current tile's 4 per-wave asyncs complete
    } else {
      WAIT_ASYNC(0);
    }
    __syncthreads();
#pragma unroll
    for (int kk = 0; kk < 32; kk += 4) {
      const int kb = kk + 2 * hf;
      v2f a;  a.x  = As[cur][(r16 + lid) * LSTR + kb];
              a.y  = As[cur][(r16 + lid) * LSTR + kb + 1];
      v2f b0; b0.x = Ws[cur][(c0 + lid) * LSTR + kb];
              b0.y = Ws[cur][(c0 + lid) * LSTR + kb + 1];
      v2f b1; b1.x = Ws[cur][(c0 + 16 + lid) * LSTR + kb];
              b1.y = Ws[cur][(c0 + 16 + lid) * LSTR + kb + 1];
      acc0 = wmma_f32(a, b0, acc0);
      acc1 = wmma_f32(a, b1, acc1);
    }
    __syncthreads();
  }

  const int e0 = n0 + c0 + lid;
  if (mode == 0) {
#pragma unroll
    for (int vi = 0; vi < 8; ++vi) {
      const int m = m0 + r16 + vi + hf * 8;
      out[(size_t)m * Nc + e0]      = acc0[vi] + bias[e0];
      out[(size_t)m * Nc + e0 + 16] = acc1[vi] + bias[e0 + 16];
    }
  } else {
#pragma unroll
    for (int vi = 0; vi < 8; ++vi) {
      const int m = m0 + r16 + vi + hf * 8;
      const int nb = m >> 11, l = m & 2047;
      {
        const int which = e0 >> 10, rem = e0 & 1023;
        const int h = rem >> 6, c = rem & 63;
        qkv_base[(size_t)which * BH_ELEMS +
                 ((size_t)((nb * 16 + h) * L_SEQ + l)) * HD + c] =
            acc0[vi] + bias[e0];
      }
      {
        const int e1 = e0 + 16;
        const int which = e1 >> 10, rem = e1 & 1023;
        const int h = rem >> 6, c = rem & 63;
        qkv_base[(size_t)which * BH_ELEMS +
                 ((size_t)((nb * 16 + h) * L_SEQ + l)) * HD + c] =
            acc1[vi] + bias[e1];
      }
    }
  }
}

// ---------------------------------------------------------------------------
// q path: y = gelu(q @ Wc^T + bc); LayerNorm(head dim); accumulate mean over L.
// One wave per row, 8 rows per block.
// ---------------------------------------------------------------------------
__global__ __launch_bounds__(256)
void qproc(const float* __restrict__ Q, const float* __restrict__ Wc,
           const float* __restrict__ bc, const float* __restrict__ gamma,
           const float* __restrict__ beta, float* __restrict__ qrel_sum)
{
  __shared__ float Wcs[64 * 64];
  __shared__ float rows[8][64];
  const int tid = threadIdx.x, lane = tid & 31, w = tid >> 5;
  for (int i = tid; i < 4096; i += 256) Wcs[i] = Wc[i];
  const size_t row = (size_t)blockIdx.x * 8 + w;
  const float* qr = Q + row * HD;
  rows[w][lane]      = qr[lane];
  rows[w][lane + 32] = qr[lane + 32];
  __syncthreads();

  const int o0 = lane, o1 = lane + 32;
  float y0 = bc[o0], y1 = bc[o1];
#pragma unroll 8
  for (int c = 0; c < 64; ++c) {
    const float qv = rows[w][c];
    y0 = fmaf(qv, Wcs[o0 * 64 + c], y0);
    y1 = fmaf(qv, Wcs[o1 * 64 + c], y1);
  }
  // exact GELU: 0.5*x*(1+erf(x/sqrt(2)))
  y0 = 0.5f * y0 * (1.0f + erff(y0 * 0.70710678118654752f));
  y1 = 0.5f * y1 * (1.0f + erff(y1 * 0.70710678118654752f));

  float s = y0 + y1, sq = y0 * y0 + y1 * y1;
#pragma unroll
  for (int m = 16; m >= 1; m >>= 1) {
    s  += __shfl_xor(s,  m, 32);
    sq += __shfl_xor(sq, m, 32);
  }
  const float mean = s * (1.0f / 64.0f);
  const float var  = sq * (1.0f / 64.0f) - mean * mean;
  const float inv  = rsqrtf(var + 1e-5f);
  const float n0v = (y0 - mean) * inv * gamma[o0] + beta[o0];
  const float n1v = (y1 - mean) * inv * gamma[o1] + beta[o1];
  const int b = (int)(row >> 11);
  atomicAdd(&qrel_sum[b * HD + o0], n0v);
  atomicAdd(&qrel_sum[b * HD + o1], n1v);
}

// s[b,l] = (qrel_sum/L * SCALING) . k[b,l,:]
__global__ void kernel_s(const float* __restrict__ Kh,
                         const float* __restrict__ qrel_sum,
                         float* __restrict__ S)
{
  const int g = blockIdx.x * 256 + threadIdx.x;  // 65536
  const int b = g >> 11;
  const float* kr = Kh + (size_t)g * HD;
  const float* q  = qrel_sum + b * HD;
  float acc = 0.f;
#pragma unroll 8
  for (int c = 0; c < 64; ++c) acc = fmaf(q[c], kr[c], acc);
  S[g] = acc * (0.125f / 2048.0f);  // SCALING=1/8, mean 1/L
}

// Per-batch: M = max(s), E[t] = exp(s[t]-M), R[i] = 1 / prefix_sum(E)[i]
__global__ __launch_bounds__(256)
void kernel_scan(const float* __restrict__ S, float* __restrict__ E,
                 float* __restrict__ R)
{
  __shared__ float red[256];
  __shared__ float Mb;
  const int b = blockIdx.x, tid = threadIdx.x;
  const float* sb = S + b * L_SEQ;
  float mx = -1e30f;
  for (int t = tid; t < L_SEQ; t += 256) mx = fmaxf(mx, sb[t]);
  red[tid] = mx; __syncthreads();
  for (int st = 128; st >= 1; st >>= 1) {
    if (tid < st) red[tid] = fmaxf(red[tid], red[tid + st]);
    __syncthreads();
  }
  if (tid == 0) Mb = red[0];
  __syncthreads();
  const float m = Mb;
  for (int t = tid; t < L_SEQ; t += 256) E[b * L_SEQ + t] = expf(sb[t] - m);
  __syncthreads();
  if (tid == 0) {
    float run = 0.f;
    for (int t = 0; t < L_SEQ; ++t) {
      run += E[b * L_SEQ + t];
      R[b * L_SEQ + t] = 1.0f / run;
    }
  }
}

// ---------------------------------------------------------------------------
// attn_out[b,i,:] = R[i] * sum_{j<=i} E[i-j] * v[b,j,:]   (Toeplitz x dense)
// WMMA fp32; V tiles via double-buffered async-LDS copies; A-fragments
// synthesized from the 1-D E window. Stores straight into (N,L,D) layout.
// ---------------------------------------------------------------------------
__global__ __launch_bounds__(256)
void toeplitz_attn(const float* __restrict__ Vh, const float* __restrict__ E,
                   const float* __restrict__ R, float* __restrict__ out)
{
  __shared__ __align__(16) float Vs[2][32 * 68];
  __shared__ float Es[96];
  const int b  = blockIdx.y;
  const int I0 = blockIdx.x * 64;
  const int tid = threadIdx.x, lane = tid & 31, w = tid >> 5;
  const int hf = lane >> 4, lid = lane & 15;
  const int r16 = (w & 3) * 16, c0 = (w >> 2) * 32;
  const float* Eb = E + b * L_SEQ;
  const float* Rb = R + b * L_SEQ;
  const float* Vb = Vh + (size_t)b * L_SEQ * HD;

  v8f acc0 = {}; v8f acc1 = {};

  const int NT = (I0 + 64) >> 5;  // number of 32-wide j chunks

  // prologue: async-fetch V tile 0 into buffer 0
  for (int f = tid; f < 512; f += 256) {
    const int rr = f >> 4, cq = (f & 15) * 4;
    async_copy_b128(lds_addr_of(&Vs[0][rr * 68 + cq]),
                    Vb + (size_t)rr * HD + cq);
  }

  for (int it = 0; it < NT; ++it) {
    const int cur = it & 1;
    const int J0 = it << 5;
    if (it + 1 < NT) {
      const int J0n = (it + 1) << 5;
      for (int f = tid; f < 512; f += 256) {
        const int rr = f >> 4, cq = (f & 15) * 4;
        async_copy_b128(lds_addr_of(&Vs[cur ^ 1][rr * 68 + cq]),
                        Vb + (size_t)(J0n + rr) * HD + cq);
      }
    }
    // E window for current tile (normal loads: needs causal zero-fill)
    const int base = I0 - J0 - 31;
    if (tid < 96) {
      const int d = base + tid;
      Es[tid] = (d >= 0 && d < L_SEQ) ? Eb[d] : 0.0f;
    }
    if (it + 1 < NT) { WAIT_ASYNC(2); } else { WAIT_ASYNC(0); }
    __syncthreads();
#pragma unroll
    for (int kk = 0; kk < 32; kk += 4) {
      const int kb = kk + 2 * hf;
      v2f a;  a.x  = Es[r16 + lid - kb + 31];
              a.y  = Es[r16 + lid - kb + 30];
      v2f b0; b0.x = Vs[cur][kb * 68 + c0 + lid];
              b0.y = Vs[cur][(kb + 1) * 68 + c0 + lid];
      v2f b1; b1.x = Vs[cur][kb * 68 + c0 + 16 + lid];
              b1.y = Vs[cur][(kb + 1) * 68 + c0 + 16 + lid];
      acc0 = wmma_f32(a, b0, acc0);
      acc1 = wmma_f32(a, b1, acc1);
    }
    __syncthreads();
  }

  const int nb = b >> 4, h = b & 15;
#pragma unroll
  for (int vi = 0; vi < 8; ++vi) {
    const int i = I0 + r16 + vi + hf * 8;
    const float sc = Rb[i];
    const size_t rowoff = ((size_t)(nb * L_SEQ + i)) * 1024 + h * HD;
    out[rowoff + c0 + lid]      = acc0[vi] * sc;
    out[rowoff + c0 + 16 + lid] = acc1[vi] * sc;
  }
}

// out2[b,j] = (1/L) * sum_{i>=j} E[i-j] * R[i]  (== weights.mean(axis=1))
__global__ void kernel_wmean(const float* __restrict__ E,
                             const float* __restrict__ R,
                             float* __restrict__ out2)
{
  const int b = blockIdx.x;
  const float* Eb = E + b * L_SEQ;
  const float* Rb = R + b * L_SEQ;
  for (int j = threadIdx.x; j < L_SEQ; j += 256) {
    float acc = 0.f;
    for (int i = j; i < L_SEQ; ++i) acc = fmaf(Eb[i - j], Rb[i], acc);
    out2[b * L_SEQ + j] = acc * (1.0f / 2048.0f);
  }
}

__global__ void kernel_zero(float* p, int n) {
  const int g = blockIdx.x * 256 + threadIdx.x;
  if (g < n) p[g] = 0.f;
}

// ---------------------------------------------------------------------------
extern "C" void kernel_launch(void* const* d_in, const int* in_sizes, int n_in,
                              void* d_out, int out_size, void* d_ws, size_t ws_size,
                              hipStream_t stream) {
  const float* x     = (const float*)d_in[0];
  const float* W_in  = (const float*)d_in[1];
  const float* b_in  = (const float*)d_in[2];
  const float* Wc    = (const float*)d_in[3];
  const float* bc    = (const float*)d_in[4];
  const float* gamma = (const float*)d_in[5];
  const float* beta  = (const float*)d_in[6];
  const float* W_out = (const float*)d_in[7];
  const float* b_out = (const float*)d_in[8];
  float* out = (float*)d_out;

  float* ws   = (float*)d_ws;
  float* qh   = ws;                    // 32*2048*64
  float* kh   = ws + BH_ELEMS;         // 32*2048*64
  float* vh   = ws + 2 * BH_ELEMS;     // 32*2048*64
  float* qrel = ws + 3 * BH_ELEMS;     // 32*64
  float* sbuf = qrel + 2048;           // 32*2048
  float* Ebuf = sbuf + 65536;          // 32*2048
  float* Rbuf = Ebuf + 65536;          // 32*2048
  float* attn = Rbuf + 65536;          // 2*2048*1024

  kernel_zero<<<8, 256, 0, stream>>>(qrel, 2048);

  // qkv = x @ W_in^T + b_in, scattered into per-head q/k/v
  gemm_xwT<<<dim3(48, 64), 256, 0, stream>>>(x, W_in, b_in, nullptr, qh,
                                             4096, 3072, 1024, 1);

  qproc<<<8192, 256, 0, stream>>>(qh, Wc, bc, gamma, beta, qrel);

  kernel_s<<<256, 256, 0, stream>>>(kh, qrel, sbuf);
  kernel_scan<<<32, 256, 0, stream>>>(sbuf, Ebuf, Rbuf);

  toeplitz_attn<<<dim3(32, 32), 256, 0, stream>>>(vh, Ebuf, Rbuf, attn);

  // output = attn_out @ W_out^T + b_out
  gemm_xwT<<<dim3(16, 64), 256, 0, stream>>>(attn, W_out, b_out, out, nullptr,
                                             4096, 1024, 1024, 0);

  kernel_wmean<<<32, 256, 0, stream>>>(Ebuf, Rbuf, out + 4194304);
}